// Dice_40810779246984
// MI455X (gfx1250) — compile-verified
//
#include <hip/hip_runtime.h>
#include <hip/hip_bf16.h>

// ---------------------------------------------------------------------------
// Dice score, MI455X (gfx1250).
// Bandwidth-bound: 134 MB streaming reads => ~5.8 us floor at 23.3 TB/s.
// pred table (16 KB) lives in LDS (async-to-LDS copy), per-pixel class
// gathers hit the 64-bank LDS. Histogram counters packed 4x u8 per u32.
// Final reduction across workgroups uses V_WMMA_F32_16X16X4_F32 with an
// all-ones B matrix (matmul-as-reduction).
// ---------------------------------------------------------------------------

typedef float v2f  __attribute__((ext_vector_type(2)));
typedef float v8f  __attribute__((ext_vector_type(8)));
typedef float f4v  __attribute__((ext_vector_type(4)));
typedef unsigned int u4v __attribute__((ext_vector_type(4)));

#define TPB   256
#define NBLK  1024   // 256k threads; 16.7M px / 4 per vec / 256k = 16 iters

// ---------------------------------------------------------------------------
// Kernel 1: pred[v] = argmax over 4 contiguous floats (first-max tie rule).
// ---------------------------------------------------------------------------
__global__ __launch_bounds__(256) void dice_argmax_kernel(
    const float* __restrict__ logits, unsigned char* __restrict__ pred, int V)
{
    int i = blockIdx.x * blockDim.x + threadIdx.x;
    if (i >= V) return;
    f4v r = ((const f4v*)logits)[i];          // row of 4 floats, 16B aligned
    int best = 0; float bv = r.x;
    if (r.y > bv) { bv = r.y; best = 1; }     // strict '>' keeps first max,
    if (r.z > bv) { bv = r.z; best = 2; }     // matching jnp.argmax ties
    if (r.w > bv) { bv = r.w; best = 3; }
    pred[i] = (unsigned char)best;
}

// ---------------------------------------------------------------------------
// Kernel 2: per-workgroup 12-bin histogram partials.
//   counters: [0..3]=eq per class, [4..7]=target count, [8..11]=pred count
// ---------------------------------------------------------------------------
__global__ __launch_bounds__(TPB) void dice_hist_kernel(
    const int* __restrict__ target, const int* __restrict__ segments,
    const unsigned char* __restrict__ pred_g, float* __restrict__ partials,
    int npix4, int V)
{
    __shared__ __align__(16) unsigned char pred_lds[16384];
    __shared__ unsigned int cnt_lds[12];

    const int tid = threadIdx.x;

    // --- async copy of the pred table (V bytes) into LDS, b128 granules ---
    {
        const unsigned char* g = pred_g;
        for (unsigned off = (unsigned)tid * 16u; off < (unsigned)V;
             off += (unsigned)TPB * 16u) {
            unsigned lds_off = (unsigned)(size_t)(const void*)(pred_lds + off);
            unsigned long long ga = (unsigned long long)(size_t)(g + off);
            asm volatile("global_load_async_to_lds_b128 %0, %1, off"
                         :: "v"(lds_off), "v"(ga) : "memory");
        }
    }
    if (tid < 12) cnt_lds[tid] = 0u;
    asm volatile("s_wait_asynccnt 0" ::: "memory");
    __syncthreads();

    // --- streaming histogram, packed counters (<=64 px/thread, fits u8) ---
    unsigned tpack = 0u, opack = 0u, eqpack = 0u;
    const int T = (int)(gridDim.x * blockDim.x);
    const u4v* __restrict__ seg4 = (const u4v*)segments;
    const u4v* __restrict__ tgt4 = (const u4v*)target;

    for (int i = blockIdx.x * TPB + tid; i < npix4; i += T) {
        u4v s = __builtin_nontemporal_load(seg4 + i);
        u4v t = __builtin_nontemporal_load(tgt4 + i);
#pragma unroll
        for (int k = 0; k < 4; ++k) {
            unsigned tv = t[k];
            unsigned ov = (unsigned)pred_lds[s[k]];
            tpack  += 1u << (tv * 8u);
            opack  += 1u << (ov * 8u);
            eqpack += (tv == ov) ? (1u << (tv * 8u)) : 0u;
        }
    }

#pragma unroll
    for (int c = 0; c < 4; ++c) {
        atomicAdd(&cnt_lds[c],      (eqpack >> (8 * c)) & 0xFFu);
        atomicAdd(&cnt_lds[4 + c],  (tpack  >> (8 * c)) & 0xFFu);
        atomicAdd(&cnt_lds[8 + c],  (opack  >> (8 * c)) & 0xFFu);
    }
    __syncthreads();

    if (tid < 12)
        partials[blockIdx.x * 12 + tid] = (float)cnt_lds[tid];
}

// ---------------------------------------------------------------------------
// Kernel 3: one wave32. Reduce G x 12 partials with V_WMMA_F32_16X16X4_F32.
//   A(16x4): row m = counter m; lane m supplies K=0,1, lane m+16 K=2,3
//   (per ISA A-matrix 16x4 layout). B = all-ones (layout independent).
//   D[m][n] = sum_k A[m][k]: lane 0 reads counters 0..7 from d[0..7],
//   lane 16 reads counters 8..11 from d[0..3] (C/D layout).
// ---------------------------------------------------------------------------
__global__ __launch_bounds__(32) void dice_finalize_kernel(
    const float* __restrict__ partials, float* __restrict__ out, int G)
{
    __shared__ float tot[16];
    const int lane = threadIdx.x;          // 0..31, one full wave32
    const int c = lane & 15;               // counter index this lane serves
    const int q = G >> 2;                  // groups per chunk
    const int chunk = (lane < 16) ? 0 : 2;

    float a0 = 0.f, a1 = 0.f;
    if (c < 12) {
        for (int g = chunk * q;       g < (chunk + 1) * q; ++g) a0 += partials[g * 12 + c];
        for (int g = (chunk + 1) * q; g < (chunk + 2) * q; ++g) a1 += partials[g * 12 + c];
    }

    v2f A; A.x = a0; A.y = a1;
    v2f B; B.x = 1.0f; B.y = 1.0f;
    v8f C = {0.f, 0.f, 0.f, 0.f, 0.f, 0.f, 0.f, 0.f};
    // D = A x ones + 0  -> every column of row m holds total of counter m
    v8f D = __builtin_amdgcn_wmma_f32_16x16x4_f32(
        false, A, false, B, (short)0, C, false, false);

    if (lane == 0) {
#pragma unroll
        for (int i = 0; i < 8; ++i) tot[i] = D[i];          // counters 0..7
    }
    if (lane == 16) {
#pragma unroll
        for (int i = 0; i < 4; ++i) tot[8 + i] = D[i];      // counters 8..11
    }
    __syncthreads();

    if (lane < 4) {
        float inter = 2.0f * tot[lane];
        float uni   = tot[4 + lane] + tot[8 + lane];
        out[lane] = inter / (uni + 1e-10f);
    }
}

// ---------------------------------------------------------------------------
extern "C" void kernel_launch(void* const* d_in, const int* in_sizes, int n_in,
                              void* d_out, int out_size, void* d_ws, size_t ws_size,
                              hipStream_t stream) {
    const float* logits   = (const float*)d_in[0];     // (V, 4) f32
    const int*   target   = (const int*)d_in[1];       // (N, N) i32
    const int*   segments = (const int*)d_in[2];       // (N, N) i32

    const int V    = in_sizes[0] / 4;                  // 16384
    const int npix = in_sizes[1];                      // 4096*4096

    unsigned char* pred = (unsigned char*)d_ws;                    // V bytes
    float* partials = (float*)((char*)d_ws + ((V + 255) & ~255));  // NBLK*12 f32
    float* out = (float*)d_out;                                    // 4 f32

    dice_argmax_kernel<<<(V + 255) / 256, 256, 0, stream>>>(logits, pred, V);
    dice_hist_kernel<<<NBLK, TPB, 0, stream>>>(target, segments, pred,
                                               partials, npix / 4, V);
    dice_finalize_kernel<<<1, 32, 0, stream>>>(partials, out, NBLK);
}